// RNN_36876589204126
// MI455X (gfx1250) — compile-verified
//
#include <hip/hip_runtime.h>

// Problem constants (from reference)
#define RNN_T    512
#define RNN_B    256
#define RNN_DIN  256
#define RNN_DLAT 512
#define RNN_DOUT 256

#define SC_BLOCKS 32   // 4 M-groups x 8 N-groups for the persistent scan

typedef __attribute__((ext_vector_type(2))) float v2f;
typedef __attribute__((ext_vector_type(8))) float v8f;

// ---------------------------------------------------------------------------
// Workgroup-tiled f32 WMMA GEMM for the two fully-parallel GEMMs.
//   C[M,N] = A[M,K] @ B[K,N] + bias[N]
// 256 threads / 8 waves per WG; WG tile = 128(M) x 64(N); K streamed in
// double-buffered 32-wide chunks through LDS (pair-major so every fragment
// is a single ds_load_b64). B panel shared by all 8 waves (8x less cache
// traffic than per-wave fetching). M,K,N must divide the tile sizes (true).
// ---------------------------------------------------------------------------
__global__ __launch_bounds__(256) void gemm_tiled_wmma(
    const float* __restrict__ A,
    const float* __restrict__ Bm,
    const float* __restrict__ bias,
    float*       __restrict__ C,
    int M, int N, int K)
{
    __shared__ float sA[2 * 4096];   // 2 x (32k x 128m) pair-major, 32KB
    __shared__ float sB[2 * 2048];   // 2 x (32k x  64n) pair-major, 16KB

    const int tid  = threadIdx.x;
    const int lane = tid & 31;
    const int wav  = tid >> 5;

    const int nTilesN = N / 64;
    const int tm = blockIdx.x / nTilesN;     // wave-uniform
    const int tn = blockIdx.x % nTilesN;
    const int m0 = tm * 128;
    const int n0 = tn * 64;

    const int wm = wav * 16;                 // wave's m-slab within WG tile

    // staging coords: A chunk = 1024 float4 (4/thread), B chunk = 512 float4 (2/thread)
    // A float4 f: m = f>>3, k4 = (f&7)*4 ; B float4 f: k = f>>4, n4 = (f&15)*4
    const int am[4] = { (tid + 0) >> 3, (tid + 256) >> 3, (tid + 512) >> 3, (tid + 768) >> 3 };
    const int ak[4] = { (tid & 7) * 4, (tid & 7) * 4, (tid & 7) * 4, (tid & 7) * 4 };
    const int bk[2] = { (tid + 0) >> 4, (tid + 256) >> 4 };
    const int bn[2] = { (tid & 15) * 4, (tid & 15) * 4 };

    // fragment coords (pair-major LDS)
    const int fk   = lane >> 4;
    const int aCol = (wm + (lane & 15)) * 2;
    const int bCol = (lane & 15) * 2;
    const int cn   = lane & 15;
    const int mh   = (lane >> 4) * 8;

    v8f acc[4];
#pragma unroll
    for (int t = 0; t < 4; ++t)
#pragma unroll
        for (int v = 0; v < 8; ++v) acc[t][v] = 0.0f;

    const int nChunks = K / 32;

    // prefetch chunk 0
    float4 ra[4], rb[2];
#pragma unroll
    for (int j = 0; j < 4; ++j)
        ra[j] = *(const float4*)(A + (size_t)(m0 + am[j]) * K + ak[j]);
#pragma unroll
    for (int j = 0; j < 2; ++j)
        rb[j] = *(const float4*)(Bm + (size_t)bk[j] * N + n0 + bn[j]);

    for (int c = 0; c < nChunks; ++c) {
        float* bufA = sA + (c & 1) * 4096;
        float* bufB = sB + (c & 1) * 2048;

        // store prefetched chunk into LDS (pair-major)
#pragma unroll
        for (int j = 0; j < 4; ++j) {
            v2f p;
            p.x = ra[j].x; p.y = ra[j].y;
            *(v2f*)&bufA[(ak[j] >> 1) * 256 + am[j] * 2]       = p;
            p.x = ra[j].z; p.y = ra[j].w;
            *(v2f*)&bufA[(ak[j] >> 1) * 256 + 256 + am[j] * 2] = p;
        }
#pragma unroll
        for (int j = 0; j < 2; ++j) {
            float* d = &bufB[(bk[j] >> 1) * 128 + bn[j] * 2 + (bk[j] & 1)];
            d[0] = rb[j].x; d[2] = rb[j].y; d[4] = rb[j].z; d[6] = rb[j].w;
        }
        __syncthreads();

        if (c + 1 < nChunks) {   // overlap next chunk's global loads with WMMA
            const int kc = (c + 1) * 32;
#pragma unroll
            for (int j = 0; j < 4; ++j)
                ra[j] = *(const float4*)(A + (size_t)(m0 + am[j]) * K + kc + ak[j]);
#pragma unroll
            for (int j = 0; j < 2; ++j)
                rb[j] = *(const float4*)(Bm + (size_t)(kc + bk[j]) * N + n0 + bn[j]);
        }

#pragma unroll
        for (int kl = 0; kl < 32; kl += 4) {
            const int row = (kl >> 1) + fk;
            v2f a = *(const v2f*)&bufA[row * 256 + aCol];
#pragma unroll
            for (int t = 0; t < 4; ++t) {
                v2f b = *(const v2f*)&bufB[row * 128 + bCol + t * 32];
                acc[t] = __builtin_amdgcn_wmma_f32_16x16x4_f32(
                    false, a, false, b, (short)0, acc[t], false, false);
            }
        }
        __syncthreads();
    }

    // epilogue: add bias, store
#pragma unroll
    for (int t = 0; t < 4; ++t) {
        const int ncol = n0 + t * 16 + cn;
        const float bs = bias[ncol];
#pragma unroll
        for (int v = 0; v < 8; ++v) {
            const size_t idx = (size_t)(m0 + wm + mh + v) * N + ncol;
            C[idx] = acc[t][v] + bs;
        }
    }
}

// ---------------------------------------------------------------------------
// Barrier counter init (d_ws is poisoned once and never re-poisoned between
// graph replays, so we must reset it deterministically every call).
// ---------------------------------------------------------------------------
__global__ void zero_ctr(unsigned* ctr) { *ctr = 0u; }

// ---------------------------------------------------------------------------
// Persistent scan kernel: for t in [0,T): hidden[t] = relu(P_t + h_{t-1}@Wr)
// (unchanged from round 2 — disasm shows well-pipelined ds_load_2addr_b64 +
//  wmma with partial dscnt waits and split barriers)
// ---------------------------------------------------------------------------
__global__ __launch_bounds__(256) void rnn_scan(
    const float* __restrict__ h0,
    const float* __restrict__ Wr,
    float*       __restrict__ hidden,
    unsigned*    __restrict__ ctr)
{
    extern __shared__ float lds[];
    float* sWr = lds;            // 32768 floats: pair-major Wr panel
    float* sA  = lds + 32768;    // 2 x 2048 floats: double-buffered A chunks

    const int tid  = threadIdx.x;
    const int lane = tid & 31;
    const int wav  = tid >> 5;

    const int mg = (blockIdx.x >> 3) * 64;   // row-group base
    const int ng = (blockIdx.x & 7) * 64;    // col-group base

    const int wm = (wav & 3) * 16;           // wave's m-slab within group
    const int wn = (wav >> 2) * 32;          // wave's n-half within group

    // ---- stage Wr panel (512 x 64) into LDS pair-major, once ----
#pragma unroll 4
    for (int i = 0; i < 32; ++i) {
        const int flat = tid + 256 * i;      // float4 id in [0,8192)
        const int k    = flat >> 4;
        const int n4   = (flat & 15) * 4;
        const float4 g = *(const float4*)(Wr + (size_t)k * RNN_DLAT + ng + n4);
        float* d = &sWr[(k >> 1) * 128 + n4 * 2 + (k & 1)];
        d[0] = g.x; d[2] = g.y; d[4] = g.z; d[6] = g.w;
    }

    // per-thread staging coordinates for A chunks (64 rows x 32 k-values)
    const int sm0 = tid >> 3;            const int sk0 = (tid & 7) * 4;
    const int sm1 = (tid + 256) >> 3;    const int sk1 = ((tid + 256) & 7) * 4;

    // fragment coordinates
    const int fm  = (wm + (lane & 15)) * 2;  // A frag LDS col (pair-major)
    const int fk  = lane >> 4;               // k-pair selector
    const int fn0 = (wn + (lane & 15)) * 2;  // B frag LDS col, tile 0
    const int cn  = lane & 15;
    const int mh  = (lane >> 4) * 8;

    __syncthreads();

    for (int t = 0; t < RNN_T; ++t) {
        const float* hprev = (t == 0) ? h0
                           : hidden + (size_t)(t - 1) * RNN_B * RNN_DLAT;
        float* ht = hidden + (size_t)t * RNN_B * RNN_DLAT;

        v8f acc[2];
#pragma unroll
        for (int q = 0; q < 2; ++q)
#pragma unroll
            for (int v = 0; v < 8; ++v) acc[q][v] = 0.0f;

        // prefetch chunk 0 of the A panel
        float4 r0 = *(const float4*)(hprev + (size_t)(mg + sm0) * RNN_DLAT + sk0);
        float4 r1 = *(const float4*)(hprev + (size_t)(mg + sm1) * RNN_DLAT + sk1);

        for (int c = 0; c < 16; ++c) {
            float* buf = sA + (c & 1) * 2048;
            {   // store prefetched chunk, pair-major (two b64 LDS stores each)
                v2f p;
                p.x = r0.x; p.y = r0.y; *(v2f*)&buf[(sk0 >> 1) * 128 + sm0 * 2]       = p;
                p.x = r0.z; p.y = r0.w; *(v2f*)&buf[(sk0 >> 1) * 128 + 128 + sm0 * 2] = p;
                p.x = r1.x; p.y = r1.y; *(v2f*)&buf[(sk1 >> 1) * 128 + sm1 * 2]       = p;
                p.x = r1.z; p.y = r1.w; *(v2f*)&buf[(sk1 >> 1) * 128 + 128 + sm1 * 2] = p;
            }
            __syncthreads();

            if (c < 15) {   // issue next chunk's global loads; overlap with WMMA
                const int kc = (c + 1) * 32;
                r0 = *(const float4*)(hprev + (size_t)(mg + sm0) * RNN_DLAT + kc + sk0);
                r1 = *(const float4*)(hprev + (size_t)(mg + sm1) * RNN_DLAT + kc + sk1);
            }

            const int kb = c * 32;
#pragma unroll
            for (int kl = 0; kl < 32; kl += 4) {
                v2f a = *(const v2f*)&buf[((kl >> 1) + fk) * 128 + fm];
                const int krow = (((kb + kl) >> 1) + fk) * 128;
                v2f b0 = *(const v2f*)&sWr[krow + fn0];
                v2f b1 = *(const v2f*)&sWr[krow + fn0 + 32];
                acc[0] = __builtin_amdgcn_wmma_f32_16x16x4_f32(
                    false, a, false, b0, (short)0, acc[0], false, false);
                acc[1] = __builtin_amdgcn_wmma_f32_16x16x4_f32(
                    false, a, false, b1, (short)0, acc[1], false, false);
            }
        }

        // epilogue: h_t = relu(acc + P_t)  (P_t lives at ht, overwritten in place)
#pragma unroll
        for (int q = 0; q < 2; ++q) {
            const int col = ng + wn + q * 16 + cn;
#pragma unroll
            for (int v = 0; v < 8; ++v) {
                const size_t idx = (size_t)(mg + wm + mh + v) * RNN_DLAT + col;
                ht[idx] = fmaxf(acc[q][v] + ht[idx], 0.0f);
            }
        }

        // ---- grid barrier between steps ----
        __threadfence();                       // publish h_t stores (release)
        __syncthreads();
        if (tid == 0) {
            __hip_atomic_fetch_add(ctr, 1u, __ATOMIC_RELEASE,
                                   __HIP_MEMORY_SCOPE_AGENT);
            const unsigned target = (unsigned)SC_BLOCKS * (unsigned)(t + 1);
            while (__hip_atomic_load(ctr, __ATOMIC_ACQUIRE,
                                     __HIP_MEMORY_SCOPE_AGENT) < target)
                __builtin_amdgcn_s_sleep(2);
        }
        __syncthreads();
        __threadfence();                       // invalidate stale lines (acquire)
    }
}

// ---------------------------------------------------------------------------
extern "C" void kernel_launch(void* const* d_in, const int* in_sizes, int n_in,
                              void* d_out, int out_size, void* d_ws, size_t ws_size,
                              hipStream_t stream) {
    (void)in_sizes; (void)n_in; (void)out_size; (void)ws_size;

    const float* x  = (const float*)d_in[0];  // (T, B, DIN)
    const float* h0 = (const float*)d_in[1];  // (B, DLAT)
    const float* Wh = (const float*)d_in[2];  // (DIN+DLAT, DLAT)
    const float* bh = (const float*)d_in[3];  // (DLAT)
    const float* Wo = (const float*)d_in[4];  // (DLAT, DOUT)
    const float* bo = (const float*)d_in[5];  // (DOUT)

    float* outputs = (float*)d_out;                                   // (T,B,DOUT)
    float* hidden  = outputs + (size_t)RNN_T * RNN_B * RNN_DOUT;      // (T,B,DLAT)
    unsigned* ctr  = (unsigned*)d_ws;

    const float* Wx = Wh;                                  // (DIN,  DLAT)
    const float* Wr = Wh + (size_t)RNN_DIN * RNN_DLAT;     // (DLAT, DLAT)

    const int MTB = RNN_T * RNN_B;   // 131072

    // ---- Kernel A: P = x @ Wx + bh  -> written straight into hidden ----
    {
        const int blocks = (MTB / 128) * (RNN_DLAT / 64);   // 8192
        gemm_tiled_wmma<<<blocks, 256, 0, stream>>>(
            x, Wx, bh, hidden, MTB, RNN_DLAT, RNN_DIN);
    }

    // ---- Persistent scan: hidden[t] = relu(P_t + hidden[t-1] @ Wr) ----
    zero_ctr<<<1, 1, 0, stream>>>(ctr);
    {
        const size_t ldsBytes = (32768 + 2 * 2048) * sizeof(float);   // 144KB
        rnn_scan<<<SC_BLOCKS, 256, ldsBytes, stream>>>(h0, Wr, hidden, ctr);
    }

    // ---- Kernel C: outputs = hidden @ Wo + bo ----
    {
        const int blocks = (MTB / 128) * (RNN_DOUT / 64);   // 4096
        gemm_tiled_wmma<<<blocks, 256, 0, stream>>>(
            hidden, Wo, bo, outputs, MTB, RNN_DOUT, RNN_DLAT);
    }
}